// PointOnSurfaceLoss_30992484008036
// MI455X (gfx1250) — compile-verified
//
#include <hip/hip_runtime.h>
#include <math.h>

typedef __attribute__((ext_vector_type(2))) float v2f;
typedef __attribute__((ext_vector_type(8))) float v8f;

#define BATCH   8
#define M_PTS   512
#define N_PTS   8192
#define EPS_F   1e-7f
#define WAVES   8
#define THREADS (WAVES * 32)
#define TILES   (N_PTS / 16)        // 512 column tiles
#define TPW     (TILES / WAVES)     // 64 tiles per wave, contiguous chunk
#define FLT_BIG 3.402823466e+38f

// D = A*B (C = 0) with the K=4 pad slot carrying the bias:
//   A[m][0..2] = -2*k_m,  A[m][3] = 1
//   B[0..2][n] = p_n,     B[3][n] = ||p_n||^2
// => D[m][n] = ||p_n||^2 - 2*k_m.p_n ; argmin_n D == argmin_n ||k_m - p_n||
__global__ __launch_bounds__(THREADS)
void pos_loss_kernel(const float* __restrict__ kp,
                     const float* __restrict__ pc,
                     const float* __restrict__ sn,
                     float* __restrict__ out)
{
    const int b    = blockIdx.y;
    const int m0   = blockIdx.x * 16;
    const int tid  = threadIdx.x;
    const int wave = tid >> 5;
    const int lane = tid & 31;
    const int col  = lane & 15;
    const bool hi  = lane >= 16;

    const float* kpb = kp + (size_t)b * 3 * M_PTS;
    const float* pcx = pc + (size_t)b * 3 * N_PTS;
    const float* pcy = pcx + N_PTS;
    const float* pcz = pcy + N_PTS;

    // A-matrix (16x4 f32): lanes 0-15 hold {K0,K1}, lanes 16-31 hold {K2,K3}.
    v2f a;
    {
        const float k0 = kpb[0 * M_PTS + m0 + col];
        const float k1 = kpb[1 * M_PTS + m0 + col];
        const float k2 = kpb[2 * M_PTS + m0 + col];
        a[0] = -2.0f * (hi ? k2 : k0);
        a[1] = hi ? 1.0f : (-2.0f * k1);   // K3 column of A = 1.0 (bias slot)
    }

    const v8f czero = {};   // loop-invariant zero accumulator

    float minv[8];
    int   mini[8];
#pragma unroll
    for (int r = 0; r < 8; ++r) { minv[r] = FLT_BIG; mini[r] = 0; }

    // Each wave sweeps a contiguous chunk of 64 tiles (compile-time trip count).
    const int t0 = wave * TPW;
#pragma unroll 4
    for (int tt = 0; tt < TPW; ++tt) {
        const int n = (t0 + tt) * 16 + col;
        const float x = pcx[n];
        const float y = pcy[n];
        const float z = pcz[n];
        const float pn2 = x * x + y * y + z * z;

        // B-matrix (4x16 f32): lanes 0-15 hold {K0,K1} = {x,y},
        //                      lanes 16-31 hold {K2,K3} = {z,||p||^2}.
        v2f bm;
        bm[0] = hi ? z : x;
        bm[1] = hi ? pn2 : y;

        v8f d = __builtin_amdgcn_wmma_f32_16x16x4_f32(
            /*neg_a=*/false, a, /*neg_b=*/false, bm,
            /*c_mod=*/(short)0, czero, /*reuse_a=*/false, /*reuse_b=*/false);

#pragma unroll
        for (int r = 0; r < 8; ++r) {
            const float v = d[r];
            if (v < minv[r]) {          // '<' keeps first index within a lane
                minv[r] = v;
                mini[r] = n;
            }
        }
    }

    // Reduce (min, idx) across the 16 lanes of each half-wave (tile columns).
#pragma unroll
    for (int r = 0; r < 8; ++r) {
#pragma unroll
        for (int off = 1; off < 16; off <<= 1) {
            const float ov = __shfl_xor(minv[r], off, 32);
            const int   oi = __shfl_xor(mini[r], off, 32);
            if (ov < minv[r] || (ov == minv[r] && oi < mini[r])) {
                minv[r] = ov;
                mini[r] = oi;
            }
        }
    }

    // D-layout: VGPR r, lanes 0-15 -> row r; lanes 16-31 -> row r+8.
    __shared__ float lv[WAVES][16];
    __shared__ int   li[WAVES][16];
    if (lane == 0 || lane == 16) {
        const int rowbase = hi ? 8 : 0;
#pragma unroll
        for (int r = 0; r < 8; ++r) {
            lv[wave][rowbase + r] = minv[r];
            li[wave][rowbase + r] = mini[r];
        }
    }
    __syncthreads();

    // Final cross-wave reduction + exact epilogue for the 16 keypoints.
    if (tid < 16) {
        const int m = tid;
        float bv = FLT_BIG;
        int   bi = 0;
#pragma unroll
        for (int w = 0; w < WAVES; ++w) {
            const float v = lv[w][m];
            const int   i = li[w][m];
            if (v < bv || (v == bv && i < bi)) { bv = v; bi = i; }
        }

        const float kx = kpb[0 * M_PTS + m0 + m];
        const float ky = kpb[1 * M_PTS + m0 + m];
        const float kz = kpb[2 * M_PTS + m0 + m];

        const float px = pcx[bi];
        const float py = pcy[bi];
        const float pz = pcz[bi];

        const float* snb = sn + (size_t)b * 3 * N_PTS;
        const float sx = snb[0 * N_PTS + bi];
        const float sy = snb[1 * N_PTS + bi];
        const float sz = snb[2 * N_PTS + bi];

        const float dx = kx - px, dy = ky - py, dz = kz - pz;
        const float dn  = sqrtf(dx * dx + dy * dy + dz * dz);
        const float inv = 1.0f / (dn + EPS_F);
        const float dot = (sx * dx + sy * dy + sz * dz) * inv;

        out[(size_t)b * M_PTS + m0 + m] = dot * dot;
    }
}

extern "C" void kernel_launch(void* const* d_in, const int* in_sizes, int n_in,
                              void* d_out, int out_size, void* d_ws, size_t ws_size,
                              hipStream_t stream) {
    const float* kp = (const float*)d_in[0];  // B x 3 x M
    const float* pc = (const float*)d_in[1];  // B x 3 x N
    const float* sn = (const float*)d_in[2];  // B x 3 x N
    float* out = (float*)d_out;               // B x M (x1x1)

    dim3 grid(M_PTS / 16, BATCH);
    pos_loss_kernel<<<grid, THREADS, 0, stream>>>(kp, pc, sn, out);
}